// SATLayer_regular_76879914598957
// MI455X (gfx1250) — compile-verified
//
#include <hip/hip_runtime.h>

#define D 64
#define TILE_STRIDE 68   // 16B-aligned rows (68*4=272B) and conflict-free columns

typedef __attribute__((ext_vector_type(2))) float v2f;
typedef __attribute__((ext_vector_type(8))) float v8f;

// ---------------------------------------------------------------------------
// Kernel 1: h = X @ W + b  (WMMA f32 16x16x4), fused a1 = h@w1+b1, a2 = h@w2+b2
// Block = 128 threads (4 waves). Block computes a 16-row x 64-col tile of h;
// wave w owns output columns [16w, 16w+16).
// ---------------------------------------------------------------------------
__global__ __launch_bounds__(128)
void gat_gemm_kernel(const float* __restrict__ X, const float* __restrict__ W,
                     const float* __restrict__ bias,
                     const float* __restrict__ w1, const float* __restrict__ b1,
                     const float* __restrict__ w2, const float* __restrict__ b2,
                     float* __restrict__ h, float* __restrict__ a1,
                     float* __restrict__ a2, int n_nodes)
{
    __shared__ float tile[16][TILE_STRIDE];

    const int row0  = blockIdx.x * 16;
    const int wave  = threadIdx.x >> 5;
    const int lane  = threadIdx.x & 31;
    const int n0    = wave * 16;        // output-column base for this wave
    const int m     = lane & 15;        // A-row / B-,C-column index
    const int khalf = (lane >> 4) * 2;  // lanes 16-31 hold K+2,K+3

    v8f c = {0.f, 0.f, 0.f, 0.f, 0.f, 0.f, 0.f, 0.f};

    const float* xrow = X + (size_t)(row0 + m) * D + khalf;

#pragma unroll
    for (int k0 = 0; k0 < D; k0 += 4) {
        v2f a, b;
        // A fragment: features[row0+m][k0+khalf .. +1]  (fuses to global_load_b64)
        a[0] = xrow[k0 + 0];
        a[1] = xrow[k0 + 1];
        // B fragment: W[k0+khalf .. +1][n0+m]   (W row-major [64][64])
        b[0] = W[(size_t)(k0 + khalf    ) * D + n0 + m];
        b[1] = W[(size_t)(k0 + khalf + 1) * D + n0 + m];
        // (neg_a, A, neg_b, B, c_mod, C, reuse_a, reuse_b)
        c = __builtin_amdgcn_wmma_f32_16x16x4_f32(false, a, false, b,
                                                  (short)0, c, false, false);
    }

    // C/D layout: VGPR r holds M = r + 8*(lane>=16), N = lane&15
    const float bn    = bias[n0 + m];
    const int   mbase = (lane >> 4) * 8;
#pragma unroll
    for (int r = 0; r < 8; ++r)
        tile[r + mbase][n0 + m] = c[r] + bn;
    __syncthreads();

    // Coalesced float4 write-back of h from the LDS tile:
    // 16 rows x 16 float4 = 256 quads; each of 128 threads stores 2.
#pragma unroll
    for (int i = 0; i < 2; ++i) {
        const int idx = threadIdx.x + i * 128;
        const int rr  = idx >> 4;        // tile row
        const int qq  = idx & 15;        // float4 index within row
        const float4 v = *(const float4*)&tile[rr][qq * 4];
        *(float4*)&h[(size_t)(row0 + rr) * D + qq * 4] = v;
    }

    // Fused attention-logit dot products: thread t < 16 handles row t.
    if (threadIdx.x < 16) {
        const int mm = threadIdx.x;
        float s1 = 0.f, s2 = 0.f;
#pragma unroll 8
        for (int n = 0; n < D; ++n) {
            const float hv = tile[mm][n];
            s1 += hv * w1[n];
            s2 += hv * w2[n];
        }
        a1[row0 + mm] = s1 + b1[0];
        a2[row0 + mm] = s2 + b2[0];
    }
}

// ---------------------------------------------------------------------------
// Kernel 2: CSR row offsets from the *sorted* edge_rows array.
// rowptr[r] = first edge index with edge_rows[e] >= r ; rowptr[N] = E.
// ---------------------------------------------------------------------------
__global__ void gat_rowptr_kernel(const int* __restrict__ rows,
                                  int* __restrict__ rowptr,
                                  int n_edges, int n_nodes)
{
    const int e = blockIdx.x * blockDim.x + threadIdx.x;
    if (e >= n_edges) return;
    const int cur  = rows[e];
    const int prev = (e == 0) ? -1 : rows[e - 1];
    for (int r = prev + 1; r <= cur; ++r) rowptr[r] = e;
    if (e == n_edges - 1)
        for (int r = cur + 1; r <= n_nodes; ++r) rowptr[r] = n_edges;
}

// ---------------------------------------------------------------------------
// Kernel 3: per-row sparse softmax + weighted aggregation.
// One wave32 per row. Lane owns output dims {2*lane, 2*lane+1} -> b64 loads.
// Online softmax fuses the max and denom passes. h gathers hit L2
// (h = 25.6 MB << 192 MB L2); prefetch hides gather latency in pass 2.
// ---------------------------------------------------------------------------
__global__ __launch_bounds__(256)
void gat_agg_kernel(const int* __restrict__ rowptr, const int* __restrict__ cols,
                    const float* __restrict__ h, const float* __restrict__ a1,
                    const float* __restrict__ a2,
                    float* __restrict__ out, int n_nodes)
{
    const int row  = blockIdx.x * 8 + (threadIdx.x >> 5);
    const int lane = threadIdx.x & 31;
    if (row >= n_nodes) return;

    const int   start = rowptr[row];
    const int   end   = rowptr[row + 1];
    const float a1r   = a1[row];

    // Pass 1: fused online softmax statistics (running max m, running sum s).
    // -FLT_MAX sentinel (not -inf) keeps exp() underflow well-defined.
    float m = -3.402823466e38f;
    float s = 0.f;
    for (int e = start + lane; e < end; e += 32) {
        const float v    = a1r + a2[cols[e]];
        const float newm = fmaxf(m, v);
        s = s * __expf(m - newm) + __expf(v - newm);
        m = newm;
    }
    // Cross-lane combine of (m, s) pairs.
    for (int off = 16; off > 0; off >>= 1) {
        const float mo   = __shfl_xor(m, off, 32);
        const float so   = __shfl_xor(s, off, 32);
        const float newm = fmaxf(m, mo);
        s = s * __expf(m - newm) + so * __expf(mo - newm);
        m = newm;
    }
    const float inv = (s > 0.f) ? 1.f / s : 0.f;

    // Pass 2: aggregate att * h[col]; all 32 lanes walk edges together,
    // one coalesced 256B (b64/lane) load of the h row per edge.
    float acc0 = 0.f, acc1 = 0.f;
    for (int e = start; e < end; ++e) {
        const int col = cols[e];
        if (e + 4 < end)  // prefetch a future edge's h row (global_prefetch_b8)
            __builtin_prefetch(h + (size_t)cols[e + 4] * D + lane * 2, 0, 3);
        const float  att = __expf(a1r + a2[col] - m) * inv;
        const float2 hv  = *(const float2*)(h + (size_t)col * D + lane * 2);
        acc0 += att * hv.x;
        acc1 += att * hv.y;
    }
    float2 res; res.x = acc0; res.y = acc1;
    *(float2*)(out + (size_t)row * D + lane * 2) = res;
}

// ---------------------------------------------------------------------------
extern "C" void kernel_launch(void* const* d_in, const int* in_sizes, int n_in,
                              void* d_out, int out_size, void* d_ws, size_t ws_size,
                              hipStream_t stream)
{
    const float* X     = (const float*)d_in[0];  // features [N, 64]
    const int*   erows = (const int*)  d_in[1];  // sorted   [E]
    const int*   ecols = (const int*)  d_in[2];  //          [E]
    const float* W     = (const float*)d_in[3];  // [64, 64]
    const float* bias  = (const float*)d_in[4];  // [64]
    const float* w1    = (const float*)d_in[5];  // [64]
    const float* b1    = (const float*)d_in[6];  // scalar
    const float* w2    = (const float*)d_in[7];  // [64]
    const float* b2    = (const float*)d_in[8];  // scalar

    const int n_nodes = in_sizes[0] / D;
    const int n_edges = in_sizes[1];

    // Workspace layout: h | a1 | a2 | rowptr   (~26.8 MB total)
    char*  ws  = (char*)d_ws;
    size_t off = 0;
    float* h      = (float*)(ws + off); off += (size_t)n_nodes * D * sizeof(float);
    float* a1     = (float*)(ws + off); off += (size_t)n_nodes * sizeof(float);
    float* a2     = (float*)(ws + off); off += (size_t)n_nodes * sizeof(float);
    int*   rowptr = (int*)  (ws + off);

    gat_gemm_kernel<<<n_nodes / 16, 128, 0, stream>>>(X, W, bias, w1, b1, w2, b2,
                                                      h, a1, a2, n_nodes);
    gat_rowptr_kernel<<<(n_edges + 255) / 256, 256, 0, stream>>>(erows, rowptr,
                                                                 n_edges, n_nodes);
    gat_agg_kernel<<<(n_nodes + 7) / 8, 256, 0, stream>>>(rowptr, ecols, h, a1, a2,
                                                          (float*)d_out, n_nodes);
}